// GCN_72602127171779
// MI455X (gfx1250) — compile-verified
//
#include <hip/hip_runtime.h>
#include <cstdint>

#define DD   64          // feature dim
#define DV   16          // DD / 4  (float4 per lane group of 16)
#define TPB  256
#define BATCH 32         // edges staged per node-group per wait

// ---------------- CDNA5 feature detection ----------------
#if defined(__has_builtin)
# if __has_builtin(__builtin_amdgcn_global_load_async_to_lds_b64)
#  define HAVE_ASYNC_LDS 1
# endif
# if __has_builtin(__builtin_amdgcn_s_wait_asynccnt)
#  define HAVE_WAIT_ASYNC 1
# endif
#endif

#if defined(HAVE_ASYNC_LDS)
#pragma message("CDNA5: global_load_async_to_lds_b64 builtin AVAILABLE (async path)")
#else
#pragma message("CDNA5: async-to-LDS builtin NOT found (wave32 shfl fallback path)")
#endif

#define AS1 __attribute__((address_space(1)))
#define AS3 __attribute__((address_space(3)))
typedef __attribute__((__vector_size__(8))) int v2i;   // int __vector(2), per hipcc diag

#if defined(HAVE_ASYNC_LDS)
__device__ __forceinline__ void wait_async0() {
#if defined(HAVE_WAIT_ASYNC)
  __builtin_amdgcn_s_wait_asynccnt(0);
#else
  asm volatile("s_wait_asynccnt 0x0" ::: "memory");
#endif
}
__device__ __forceinline__ void async_edge(const int2* g, int2* l) {
  __builtin_amdgcn_global_load_async_to_lds_b64(
      (AS1 v2i*)(uintptr_t)(const void*)g,
      (AS3 v2i*)(uint32_t)(uintptr_t)(void*)l, 0, 0);
}
#endif

// ---------------- fused SpMM + tanh + acc (hot kernel) ----------------
// 16 lanes per node, each lane owns one float4 (16 B) of the 256 B row.
// layer_new[n] = tanh( sum_{p in row n} val[p] * layerIn[col[p]] ); acc += layer_new.
__global__ __launch_bounds__(TPB) void spmm_tanh_acc(
    const float4* __restrict__ layerIn,
    const int2*  __restrict__ edges,
    const int*   __restrict__ rowPtr,
    float4* __restrict__ layerOut,
    float4* __restrict__ accOut,
    int nNodes)
{
  const int lane  = threadIdx.x & 15;
  const int group = threadIdx.x >> 4;            // 0..15
  const int node  = blockIdx.x * 16 + group;
#if defined(HAVE_ASYNC_LDS)
  __shared__ int2 stage[16 * BATCH];             // 16 groups x 32 edges x 8 B = 4 KB
  int2* gs = &stage[group * BATCH];
#endif
  if (node >= nNodes) return;

  int p = rowPtr[node];
  const int pend = rowPtr[node + 1];
  float4 acc = make_float4(0.f, 0.f, 0.f, 0.f);

  while (p < pend) {
    int n = pend - p; n = (n > BATCH) ? BATCH : n;
#if defined(HAVE_ASYNC_LDS)
    // Stage this group's next BATCH (col,val) pairs straight into LDS via the
    // CDNA5 async copy engine (ASYNCcnt), one b64 per active lane per issue.
    if (lane < n)      async_edge(edges + p + lane,      gs + lane);
    if (lane + 16 < n) async_edge(edges + p + lane + 16, gs + lane + 16);
    if (p + BATCH < pend) __builtin_prefetch(edges + p + BATCH, 0, 1);
    wait_async0();
    int j = 0;
    // 4-wide software pipeline: 4 independent ds broadcasts -> 4 independent
    // global_load_b128 gathers in flight per lane before the FMA chain waits.
    for (; j + 4 <= n; j += 4) {
      int2 e0 = gs[j], e1 = gs[j + 1], e2 = gs[j + 2], e3 = gs[j + 3];
      float4 s0 = layerIn[(size_t)(unsigned)e0.x * DV + lane];
      float4 s1 = layerIn[(size_t)(unsigned)e1.x * DV + lane];
      float4 s2 = layerIn[(size_t)(unsigned)e2.x * DV + lane];
      float4 s3 = layerIn[(size_t)(unsigned)e3.x * DV + lane];
      float w0 = __int_as_float(e0.y), w1 = __int_as_float(e1.y);
      float w2 = __int_as_float(e2.y), w3 = __int_as_float(e3.y);
      acc.x += w0 * s0.x; acc.y += w0 * s0.y; acc.z += w0 * s0.z; acc.w += w0 * s0.w;
      acc.x += w1 * s1.x; acc.y += w1 * s1.y; acc.z += w1 * s1.z; acc.w += w1 * s1.w;
      acc.x += w2 * s2.x; acc.y += w2 * s2.y; acc.z += w2 * s2.z; acc.w += w2 * s2.w;
      acc.x += w3 * s3.x; acc.y += w3 * s3.y; acc.z += w3 * s3.z; acc.w += w3 * s3.w;
    }
    for (; j < n; ++j) {
      int2  e  = gs[j];
      float wv = __int_as_float(e.y);
      float4 s = layerIn[(size_t)(unsigned)e.x * DV + lane];
      acc.x += wv * s.x; acc.y += wv * s.y; acc.z += wv * s.z; acc.w += wv * s.w;
    }
#else
    {
      int nn = (n > 16) ? 16 : n;                // register-staged fallback: 16/batch
      int idx = (lane < nn) ? lane : (nn - 1);
      int2 e  = edges[p + idx];
      const int base = (group & 1) * 16;         // this group's half of the wave32
      for (int j = 0; j < nn; ++j) {
        int   c  = __shfl(e.x, base + j, 32);
        float wv = __int_as_float(__shfl(e.y, base + j, 32));
        float4 s = layerIn[(size_t)(unsigned)c * DV + lane];
        acc.x += wv * s.x; acc.y += wv * s.y; acc.z += wv * s.z; acc.w += wv * s.w;
      }
      n = nn;
    }
#endif
    p += n;
  }

  float4 t;
  t.x = tanhf(acc.x); t.y = tanhf(acc.y); t.z = tanhf(acc.z); t.w = tanhf(acc.w);
  size_t o = (size_t)node * DV + lane;
  layerOut[o] = t;                               // next layer's input
  float4 a = accOut[o];                          // single-owner RMW, no atomics
  a.x += t.x; a.y += t.y; a.z += t.z; a.w += t.w;
  accOut[o] = a;
}

// ---------------- CSR build ----------------
__global__ __launch_bounds__(TPB) void k_zero(int* __restrict__ p, int n) {
  int i = blockIdx.x * TPB + threadIdx.x;
  if (i < n) p[i] = 0;
}

__global__ __launch_bounds__(TPB) void k_count(const int* __restrict__ rows,
                                               int* __restrict__ cnt, int E) {
  int e = blockIdx.x * TPB + threadIdx.x;
  if (e < E) atomicAdd(&cnt[rows[e]], 1);
}

// single-block exclusive scan of `cnt[0..n)` -> rowPtr[0..n], copy to rowFill
__global__ __launch_bounds__(1024) void k_scan(const int* __restrict__ cnt,
                                               int* __restrict__ rowPtr,
                                               int* __restrict__ rowFill, int n) {
  __shared__ int sums[1024];
  const int t = threadIdx.x;
  const int chunk = (n + 1023) / 1024;
  const int begin = t * chunk;
  const int end   = (begin + chunk < n) ? (begin + chunk) : n;
  int s = 0;
  for (int i = begin; i < end; ++i) s += cnt[i];
  sums[t] = s;
  __syncthreads();
  for (int off = 1; off < 1024; off <<= 1) {
    int v = (t >= off) ? sums[t - off] : 0;
    __syncthreads();
    sums[t] += v;
    __syncthreads();
  }
  int prefix = (t == 0) ? 0 : sums[t - 1];
  for (int i = begin; i < end; ++i) {
    rowPtr[i]  = prefix;
    rowFill[i] = prefix;
    prefix += cnt[i];
  }
  if (end == n) rowPtr[n] = prefix;   // rowPtr[n] = E (all qualifying threads write E)
}

__global__ __launch_bounds__(TPB) void k_scatter(const int* __restrict__ rows,
                                                 const int* __restrict__ cols,
                                                 const float* __restrict__ vals,
                                                 int* __restrict__ rowFill,
                                                 int2* __restrict__ edges, int E) {
  int e = blockIdx.x * TPB + threadIdx.x;
  if (e < E) {
    int pos = atomicAdd(&rowFill[rows[e]], 1);
    edges[pos] = make_int2(cols[e], __float_as_int(vals[e]));
  }
}

// out = weight (rows 0..N), layerA = weight[1:]
__global__ __launch_bounds__(TPB) void k_init(const float* __restrict__ w,
                                              float* __restrict__ out,
                                              float* __restrict__ layerA,
                                              size_t total) {
  size_t i = (size_t)blockIdx.x * TPB + threadIdx.x;
  if (i < total) {
    float v = w[i];
    out[i] = v;
    if (i >= DD) layerA[i - DD] = v;
  }
}

// ---------------- host ----------------
extern "C" void kernel_launch(void* const* d_in, const int* in_sizes, int n_in,
                              void* d_out, int out_size, void* d_ws, size_t ws_size,
                              hipStream_t stream) {
  const float* weight = (const float*)d_in[0];
  const int*   rows   = (const int*)d_in[1];
  const int*   cols   = (const int*)d_in[2];
  const float* vals   = (const float*)d_in[3];

  const int N = in_sizes[0] / DD - 1;   // 100000
  const int E = in_sizes[1];            // 1600000

  auto al = [](size_t x) { return (x + 255) & ~(size_t)255; };
  char* w = (char*)d_ws;
  const size_t szLayer = al((size_t)N * DD * sizeof(float));
  const size_t szCnt   = al((size_t)(N + 1) * sizeof(int));
  const size_t szEdge  = al((size_t)E * sizeof(int2));
  float* layerA  = (float*)w;  w += szLayer;
  float* layerB  = (float*)w;  w += szLayer;
  int*   cnt     = (int*)w;    w += szCnt;
  int*   rowPtr  = (int*)w;    w += szCnt;
  int*   rowFill = (int*)w;    w += szCnt;
  int2*  edges   = (int2*)w;   w += szEdge;

  // CSR build (once; reused by all 3 layers)
  hipLaunchKernelGGL(k_zero,    dim3((N + TPB - 1) / TPB), dim3(TPB), 0, stream, cnt, N);
  hipLaunchKernelGGL(k_count,   dim3((E + TPB - 1) / TPB), dim3(TPB), 0, stream, rows, cnt, E);
  hipLaunchKernelGGL(k_scan,    dim3(1), dim3(1024), 0, stream, cnt, rowPtr, rowFill, N);
  hipLaunchKernelGGL(k_scatter, dim3((E + TPB - 1) / TPB), dim3(TPB), 0, stream,
                     rows, cols, vals, rowFill, edges, E);

  // out = weight; layerA = weight[1:]
  const size_t total = (size_t)(N + 1) * DD;
  hipLaunchKernelGGL(k_init, dim3((unsigned)((total + TPB - 1) / TPB)), dim3(TPB), 0, stream,
                     weight, (float*)d_out, layerA, total);

  float4* accOut = (float4*)((float*)d_out + DD);   // rows 1..N of the output
  float* cur = layerA;
  float* nxt = layerB;
  const int spmmBlocks = (N + 15) / 16;
  for (int l = 0; l < 3; ++l) {
    hipLaunchKernelGGL(spmm_tanh_acc, dim3(spmmBlocks), dim3(TPB), 0, stream,
                       (const float4*)cur, edges, rowPtr, (float4*)nxt, accOut, N);
    float* t = cur; cur = nxt; nxt = t;
  }
}